// H2GCNConv_87342454931706
// MI455X (gfx1250) — compile-verified
//
#include <hip/hip_runtime.h>

// ---------------------------------------------------------------------------
// H2GCN conv on MI455X: out[:,0:256] = scatter(val1*x[col1] -> row1),
// out[:,256:512] = scatter(val2*x[col2] -> row2).  N=50000, D=256,
// E1=800k, E2=1.6M.  Memory-bound (~1.2 GFLOP vs ~2.6 GB of L2 traffic;
// x=51MB and out=102MB both fit MI455X's 192MB L2).
//
// Primary path: device-side CSR build (count -> scan -> scatter), then a
// row-parallel gather where one wave32 owns one output row and the next
// edge's 1KB feature row is DMA-staged into an LDS double buffer with
// global_load_async_to_lds_b128 (ASYNCcnt) while the current row is FMA'd
// out of LDS.  No f32 atomics anywhere in this path.
// Fallback (ws too small): edge-parallel global_atomic_add_f32 scatter.
// ---------------------------------------------------------------------------

#define D_FEAT     256
#define OUT_STRIDE 512
#define LANES      32
#define SCAN_BLK   1024
#define WAVES_PER_BLK 8          // 256 threads

// ---------------------------------------------------------------- utilities
__global__ void h2gcn_zero_f4(float4* __restrict__ p, long n4) {
    long i = (long)blockIdx.x * blockDim.x + threadIdx.x;
    long s = (long)gridDim.x * blockDim.x;
    const float4 z = {0.f, 0.f, 0.f, 0.f};
    for (; i < n4; i += s) p[i] = z;
}

__global__ void h2gcn_zero_i32(int* __restrict__ p, long n) {
    long i = (long)blockIdx.x * blockDim.x + threadIdx.x;
    long s = (long)gridDim.x * blockDim.x;
    for (; i < n; i += s) p[i] = 0;
}

// ------------------------------------------------------------- CSR build
__global__ void h2gcn_count_kernel(const int* __restrict__ row, int nE,
                                   int* __restrict__ cnt) {
    long i = (long)blockIdx.x * blockDim.x + threadIdx.x;
    long s = (long)gridDim.x * blockDim.x;
    for (; i < nE; i += s) atomicAdd(&cnt[row[i]], 1);
}

// exclusive prefix sum over n counts, single 1024-thread block.
__global__ void h2gcn_scan_kernel(const int* __restrict__ cnt, int n, int nE,
                                  int* __restrict__ off, int* __restrict__ cur) {
    __shared__ int sums[SCAN_BLK];
    const int t     = threadIdx.x;
    const int chunk = (n + SCAN_BLK - 1) / SCAN_BLK;
    const int lo    = t * chunk;
    const int hi    = min(n, lo + chunk);

    int s = 0;
    for (int i = lo; i < hi; ++i) s += cnt[i];
    sums[t] = s;
    __syncthreads();

    if (t == 0) {
        int run = 0;
        for (int i = 0; i < SCAN_BLK; ++i) { int v = sums[i]; sums[i] = run; run += v; }
    }
    __syncthreads();

    int run = sums[t];
    for (int i = lo; i < hi; ++i) {
        int v  = cnt[i];       // read before possible alias write
        off[i] = run;
        cur[i] = run;
        run += v;
    }
    if (t == 0) off[n] = nE;
}

__global__ void h2gcn_scatter_kernel(const int* __restrict__ row,
                                     const int* __restrict__ col,
                                     const float* __restrict__ val, int nE,
                                     int* __restrict__ cur,
                                     int* __restrict__ scol,
                                     float* __restrict__ sval) {
    long i = (long)blockIdx.x * blockDim.x + threadIdx.x;
    long s = (long)gridDim.x * blockDim.x;
    for (; i < nE; i += s) {
        int pos   = atomicAdd(&cur[row[i]], 1);
        scol[pos] = col[i];
        sval[pos] = val[i];
    }
}

// ---------------------------------------------- async-staged row gather
// Stage one 1KB feature row (256 f32) into LDS with two wave-wide
// global_load_async_to_lds_b128 ops (32 lanes x 16B each = 512B per op).
// s_wait_dscnt 0 first: the buffer being overwritten must have had its
// previous ds_load reads retired into VGPRs (ISA: DScnt==0 => data in VGPRs).
__device__ __forceinline__ void async_stage_row(unsigned ldsBase,
                                                const float* __restrict__ rowPtr,
                                                int lane) {
    const float* g0 = rowPtr + lane * 4;          // features [0,128)
    const float* g1 = rowPtr + 128 + lane * 4;    // features [128,256)
    unsigned     l0 = ldsBase + lane * 16;
    unsigned     l1 = ldsBase + 512 + lane * 16;
    asm volatile("s_wait_dscnt 0x0\n\t"
                 "global_load_async_to_lds_b128 %0, %2, off\n\t"
                 "global_load_async_to_lds_b128 %1, %3, off"
                 :: "v"(l0), "v"(l1), "v"(g0), "v"(g1)
                 : "memory");
}
// Async loads complete in order: waiting ASYNCcnt<=2 after issuing the next
// row's 2 ops guarantees the current row's 2 ops have landed in LDS.
__device__ __forceinline__ void wait_async_le2() {
    asm volatile("s_wait_asynccnt 0x2" ::: "memory");
}
__device__ __forceinline__ void wait_async_le0() {
    asm volatile("s_wait_asynccnt 0x0" ::: "memory");
}

// One wave32 per output row.  Each lane owns 8 features (2x float4).
// Double-buffered LDS staging of x[col] rows; accumulate in VGPRs; single
// pair of plain b128 stores per row (deg-0 rows store zeros, so no separate
// output clear is needed).  All control flow is wave-uniform -> EXEC is
// all-ones at every async issue, as the async ops require.
__global__ void __launch_bounds__(256)
h2gcn_row_gather_kernel(const float* __restrict__ x,
                        const int*   __restrict__ off,
                        const int*   __restrict__ scol,
                        const float* __restrict__ sval,
                        int n, float* __restrict__ out,
                        int colOff) {
    __shared__ __align__(16) float stage[WAVES_PER_BLK][2][D_FEAT]; // 16KB/block

    const int  lane   = threadIdx.x & (LANES - 1);
    const int  wid    = threadIdx.x >> 5;
    long       wave   = ((long)blockIdx.x * blockDim.x + threadIdx.x) >> 5;
    const long nWaves = ((long)gridDim.x * blockDim.x) >> 5;

    // wave-relative LDS byte offset of this wave's buffer pair (flat shared
    // pointers carry the LDS offset in their low 32 bits on gfx1250)
    const unsigned ldsBase0 = (unsigned)(unsigned long long)&stage[wid][0][0];
    const unsigned ldsBase1 = (unsigned)(unsigned long long)&stage[wid][1][0];

    for (long r = wave; r < n; r += nWaves) {
        const int beg = off[r];
        const int end = off[r + 1];

        float4 a = {0.f, 0.f, 0.f, 0.f};
        float4 b = {0.f, 0.f, 0.f, 0.f};

        if (end > beg) {
            // prologue: stage first edge's row into buffer 0
            async_stage_row(ldsBase0, x + (size_t)scol[beg] * D_FEAT, lane);

            for (int k = beg; k < end; ++k) {
                const unsigned bufCur = ((k - beg) & 1) ? ldsBase1 : ldsBase0;
                const unsigned bufNxt = ((k - beg) & 1) ? ldsBase0 : ldsBase1;

                if (k + 1 < end) {
                    async_stage_row(bufNxt, x + (size_t)scol[k + 1] * D_FEAT, lane);
                    wait_async_le2();      // current buffer resident
                } else {
                    wait_async_le0();      // last edge: drain
                }

                const float v = sval[k];
                const float4* __restrict__ lsrc =
                    (const float4*)&stage[wid][(k - beg) & 1][0];
                const float4 xa = lsrc[lane];           // ds_load_b128
                const float4 xb = lsrc[lane + LANES];   // ds_load_b128
                (void)bufCur;

                a.x = fmaf(v, xa.x, a.x); a.y = fmaf(v, xa.y, a.y);
                a.z = fmaf(v, xa.z, a.z); a.w = fmaf(v, xa.w, a.w);
                b.x = fmaf(v, xb.x, b.x); b.y = fmaf(v, xb.y, b.y);
                b.z = fmaf(v, xb.z, b.z); b.w = fmaf(v, xb.w, b.w);
            }
        }

        float4* __restrict__ dst = (float4*)(out + (size_t)r * OUT_STRIDE + colOff);
        dst[lane]         = a;
        dst[lane + LANES] = b;
    }
}

// ------------------------------------------- fallback: edge-parallel atomics
__device__ __forceinline__ void atom_add_f32(float* p, float v) {
    asm volatile("global_atomic_add_f32 %0, %1, off scope:SCOPE_DEV"
                 :: "v"(p), "v"(v)
                 : "memory");
}

__global__ void h2gcn_spmm_atomic_kernel(const float* __restrict__ x,
                                         const int*   __restrict__ row,
                                         const int*   __restrict__ col,
                                         const float* __restrict__ val,
                                         int nE, float* __restrict__ out,
                                         int colOff) {
    const int  lane   = threadIdx.x & (LANES - 1);
    long       wave   = ((long)blockIdx.x * blockDim.x + threadIdx.x) >> 5;
    const long nWaves = ((long)gridDim.x * blockDim.x) >> 5;

    for (long e = wave; e < nE; e += nWaves) {
        const int   r = row[e];
        const int   c = col[e];
        const float v = val[e];

        const long en = e + nWaves;
        if (en < nE) {
            const int cn = col[en];
            __builtin_prefetch(x + (size_t)cn * D_FEAT + (size_t)lane * 8, 0, 1);
        }

        const float4* __restrict__ src = (const float4*)(x + (size_t)c * D_FEAT);
        float4 a = src[lane];
        float4 b = src[lane + LANES];
        a.x *= v; a.y *= v; a.z *= v; a.w *= v;
        b.x *= v; b.y *= v; b.z *= v; b.w *= v;

        float* __restrict__ dst = out + (size_t)r * OUT_STRIDE + colOff;
        const int o = lane * 4;
        atom_add_f32(dst + o + 0, a.x);
        atom_add_f32(dst + o + 1, a.y);
        atom_add_f32(dst + o + 2, a.z);
        atom_add_f32(dst + o + 3, a.w);
        atom_add_f32(dst + 128 + o + 0, b.x);
        atom_add_f32(dst + 128 + o + 1, b.y);
        atom_add_f32(dst + 128 + o + 2, b.z);
        atom_add_f32(dst + 128 + o + 3, b.w);
    }
}

// --------------------------------------------------------------- launcher
extern "C" void kernel_launch(void* const* d_in, const int* in_sizes, int n_in,
                              void* d_out, int out_size, void* d_ws, size_t ws_size,
                              hipStream_t stream) {
    // setup_inputs() order: x, row1, col1, val1, row2, col2, val2
    const float* x    = (const float*)d_in[0];
    const int*   row1 = (const int*)  d_in[1];
    const int*   col1 = (const int*)  d_in[2];
    const float* val1 = (const float*)d_in[3];
    const int*   row2 = (const int*)  d_in[4];
    const int*   col2 = (const int*)  d_in[5];
    const float* val2 = (const float*)d_in[6];
    float*       out  = (float*)d_out;

    const int nE1 = in_sizes[1];                  // 800000
    const int nE2 = in_sizes[4];                  // 1600000
    const int n   = in_sizes[0] / D_FEAT;         // 50000 nodes

    // Workspace layout (4B elems):
    //   off1[n+1] cur1[n] off2[n+1] cur2[n] scol1[E1] sval1[E1] scol2[E2] sval2[E2]
    const size_t need = (size_t)4 * ((size_t)2 * (2 * (size_t)n + 1)
                                     + 2 * ((size_t)nE1 + (size_t)nE2));

    if (ws_size >= need) {
        // ---------------- CSR path (no f32 atomics) ----------------
        int* w     = (int*)d_ws;
        int* off1  = w;                 w += n + 1;
        int* cur1  = w;                 w += n;
        int* off2  = w;                 w += n + 1;
        int* cur2  = w;                 w += n;
        int*   scol1 = w;               w += nE1;
        float* sval1 = (float*)w;       w += nE1;
        int*   scol2 = w;               w += nE2;
        float* sval2 = (float*)w;       w += nE2;

        const int edgeBlk1 = (int)min((long)((nE1 + 255) / 256), 4096L);
        const int edgeBlk2 = (int)min((long)((nE2 + 255) / 256), 4096L);
        const int rowBlks  = (n * LANES + 255) / 256;   // one wave per row

        // hop 1
        h2gcn_zero_i32<<<256, 256, 0, stream>>>(cur1, n);
        h2gcn_count_kernel<<<edgeBlk1, 256, 0, stream>>>(row1, nE1, cur1);
        h2gcn_scan_kernel<<<1, SCAN_BLK, 0, stream>>>(cur1, n, nE1, off1, cur1);
        h2gcn_scatter_kernel<<<edgeBlk1, 256, 0, stream>>>(row1, col1, val1, nE1,
                                                           cur1, scol1, sval1);
        // hop 2
        h2gcn_zero_i32<<<256, 256, 0, stream>>>(cur2, n);
        h2gcn_count_kernel<<<edgeBlk2, 256, 0, stream>>>(row2, nE2, cur2);
        h2gcn_scan_kernel<<<1, SCAN_BLK, 0, stream>>>(cur2, n, nE2, off2, cur2);
        h2gcn_scatter_kernel<<<edgeBlk2, 256, 0, stream>>>(row2, col2, val2, nE2,
                                                           cur2, scol2, sval2);

        // async-staged gather (writes every output element)
        h2gcn_row_gather_kernel<<<rowBlks, 256, 0, stream>>>(x, off1, scol1, sval1,
                                                             n, out, 0);
        h2gcn_row_gather_kernel<<<rowBlks, 256, 0, stream>>>(x, off2, scol2, sval2,
                                                             n, out, D_FEAT);
    } else {
        // ---------------- fallback: atomic scatter ----------------
        const long n4 = (long)out_size / 4;
        h2gcn_zero_f4<<<4096, 256, 0, stream>>>((float4*)out, n4);
        h2gcn_spmm_atomic_kernel<<<8192, 256, 0, stream>>>(x, row1, col1, val1,
                                                           nE1, out, 0);
        h2gcn_spmm_atomic_kernel<<<8192, 256, 0, stream>>>(x, row2, col2, val2,
                                                           nE2, out, D_FEAT);
    }
}